// PositiveOrientationPlaquetteGenerator_20023137534662
// MI455X (gfx1250) — compile-verified
//
#include <hip/hip_runtime.h>
#include <stdint.h>

// Plaquette P_{mu,nu}(x) = U_mu(x) U_nu(x+mu) U_mu^dag(x+nu) U_nu^dag(x), nu<mu.
// Lattice (T,X,Y,Z)=(48,24,24,24), SU(3), planar re/im inputs, interleaved
// complex64 output [6,V,3,3].
//
// Bandwidth-bound (478 MB compulsory, ~2.6 GFLOP -> ~20us floor @ 23.3 TB/s):
//  - fp32 VALU FMA chains (dual-issued) for the 3x3 complex matmuls
//  - CDNA5 async staging: global_load_async_to_lds_b128 + s_wait_asynccnt
//    shares y/z-neighbor links across the block via LDS
//  - non-temporal stores keep the 287 MB output stream from evicting the
//    191 MB gauge-link working set out of the 192 MB L2
//  - global_prefetch_b8 overlaps t/x-neighbor fetches with the async fill

#define T_DIM 48
#define X_DIM 24
#define Y_DIM 24
#define Z_DIM 24
#define VSITES (T_DIM * X_DIM * Y_DIM * Z_DIM)   // 663552
#define YTILE 8
#define NTHREADS 192                              // YTILE * Z_DIM
#define NROWS (YTILE + 1)                         // tile + y+1 halo
// LDS: [part(re/im)][mu][row][z][9] floats
#define LDS_MATS (4 * NROWS * Z_DIM)              // 864 matrices per part
#define LDS_PART_FLOATS (LDS_MATS * 9)            // 7776
#define LDS_FLOATS (2 * LDS_PART_FLOATS)          // 15552 floats = 62208 B
// staging: chunk = one z-line of one (part,mu,row): 24 sites * 9 floats = 864 B
#define CHUNK_BYTES 864
#define CHUNK_B128 54
#define NCHUNKS (2 * 4 * NROWS)                   // 72
#define NXFERS (NCHUNKS * CHUNK_B128)             // 3888 b128 transfers

typedef float v2f __attribute__((ext_vector_type(2)));

struct C33 { float r[9]; float i[9]; };

__device__ __forceinline__ void gmat(C33& M, const float* __restrict__ Ure,
                                     const float* __restrict__ Uim, size_t idx9) {
  const float* pr = Ure + idx9;
  const float* pi = Uim + idx9;
#pragma unroll
  for (int k = 0; k < 9; ++k) { M.r[k] = pr[k]; M.i[k] = pi[k]; }
}

__device__ __forceinline__ void lmat(C33& M, const float* lds, int mi) {
  const float* pr = lds + mi * 9;
  const float* pi = lds + LDS_PART_FLOATS + mi * 9;
#pragma unroll
  for (int k = 0; k < 9; ++k) { M.r[k] = pr[k]; M.i[k] = pi[k]; }
}

// C = A * B   (complex 3x3)
__device__ __forceinline__ void mmul(C33& C, const C33& A, const C33& B) {
#pragma unroll
  for (int ii = 0; ii < 3; ++ii) {
#pragma unroll
    for (int jj = 0; jj < 3; ++jj) {
      float cr = 0.f, ci = 0.f;
#pragma unroll
      for (int k = 0; k < 3; ++k) {
        float ar = A.r[ii * 3 + k], ai = A.i[ii * 3 + k];
        float br = B.r[k * 3 + jj], bi = B.i[k * 3 + jj];
        cr = fmaf(ar, br, cr); cr = fmaf(-ai, bi, cr);
        ci = fmaf(ar, bi, ci); ci = fmaf(ai, br, ci);
      }
      C.r[ii * 3 + jj] = cr; C.i[ii * 3 + jj] = ci;
    }
  }
}

// C = A * B^dagger :  C_ij = sum_k A_ik * conj(B_jk)
__device__ __forceinline__ void mmul_dag(C33& C, const C33& A, const C33& B) {
#pragma unroll
  for (int ii = 0; ii < 3; ++ii) {
#pragma unroll
    for (int jj = 0; jj < 3; ++jj) {
      float cr = 0.f, ci = 0.f;
#pragma unroll
      for (int k = 0; k < 3; ++k) {
        float ar = A.r[ii * 3 + k], ai = A.i[ii * 3 + k];
        float br = B.r[jj * 3 + k], bi = B.i[jj * 3 + k];
        cr = fmaf(ar, br, cr); cr = fmaf(ai, bi, cr);
        ci = fmaf(ai, br, ci); ci = fmaf(-ar, bi, ci);
      }
      C.r[ii * 3 + jj] = cr; C.i[ii * 3 + jj] = ci;
    }
  }
}

// Non-temporal interleaved store: output is write-once streaming; NT keeps it
// from evicting the gauge links out of L2.
__device__ __forceinline__ void store_mat(float* __restrict__ p, const C33& M) {
#pragma unroll
  for (int k = 0; k < 9; ++k) {
    v2f v = {M.r[k], M.i[k]};
    __builtin_nontemporal_store(v, reinterpret_cast<v2f*>(p) + k);
  }
}

__global__ __launch_bounds__(NTHREADS)
void plaquette_kernel(const float* __restrict__ Ure,
                      const float* __restrict__ Uim,
                      float* __restrict__ out) {
  __shared__ float lds[LDS_FLOATS];

  const int tid = threadIdx.x;
  const int bid = blockIdx.x;
  const int ytile = bid % (Y_DIM / YTILE);
  const int x = (bid / (Y_DIM / YTILE)) % X_DIM;
  const int t = bid / ((Y_DIM / YTILE) * X_DIM);
  const int y0 = ytile * YTILE;

  // ---- async stage: U_mu(re/im) for rows y0..y0+YTILE (halo), full z-line ----
  const unsigned ldsBase = (unsigned)(uintptr_t)(void*)&lds[0]; // low 32 bits = wave-relative LDS offset
  for (int it = tid; it < NXFERS; it += NTHREADS) {
    const int chunk = it / CHUNK_B128;      // 0..NCHUNKS-1
    const int off16 = it % CHUNK_B128;      // 16B units in chunk
    const int part  = chunk / (4 * NROWS);  // 0=re, 1=im
    const int rem   = chunk % (4 * NROWS);
    const int mu    = rem / NROWS;
    int yy = y0 + (rem % NROWS);
    if (yy >= Y_DIM) yy -= Y_DIM;           // periodic y halo
    const float* src = part ? Uim : Ure;
    const size_t gfl = ((((size_t)mu * T_DIM + t) * X_DIM + x) * Y_DIM + yy) * (size_t)(Z_DIM * 9);
    const uint64_t gaddr = (uint64_t)(uintptr_t)(src + gfl) + (uint64_t)off16 * 16u;
    const unsigned laddr = ldsBase + (unsigned)(chunk * CHUNK_BYTES + off16 * 16);
    asm volatile("global_load_async_to_lds_b128 %0, %1, off"
                 :: "v"(laddr), "v"(gaddr) : "memory");
  }

  // ---- site / neighbor indexing ---------------------------------------------
  const int yl = tid / Z_DIM;               // 0..YTILE-1 local y row
  const int z  = tid % Z_DIM;
  const int y  = y0 + yl;
  const int zp = (z + 1 == Z_DIM) ? 0 : z + 1;
  const int tn = (t + 1 == T_DIM) ? 0 : t + 1;
  const int xn = (x + 1 == X_DIM) ? 0 : x + 1;

  const size_t s  = ((((size_t)t * X_DIM + x) * Y_DIM + y) * Z_DIM) + z;
  const size_t sT = ((((size_t)tn * X_DIM + x) * Y_DIM + y) * Z_DIM) + z;   // x+e_t
  const size_t sX = ((((size_t)t * X_DIM + xn) * Y_DIM + y) * Z_DIM) + z;   // x+e_x

#define MI(mu, row, zz) (((mu) * NROWS + (row)) * Z_DIM + (zz))
#define GIDX(mu, site) (((size_t)(mu) * VSITES + (site)) * 9)

  C33 F1, F2, F3, F4, T1, T2;

  // Overlap with the async fill: pair 0's global neighbor links now,
  // prefetch the remaining t/x neighbor links for pairs 1-4.
  gmat(F2, Ure, Uim, GIDX(0, sX));          // U_0(x+e_x)
  gmat(F3, Ure, Uim, GIDX(1, sT));          // U_1(x+e_t)
  __builtin_prefetch(Ure + GIDX(2, sT), 0, 3);
  __builtin_prefetch(Uim + GIDX(2, sT), 0, 3);
  __builtin_prefetch(Ure + GIDX(2, sX), 0, 3);
  __builtin_prefetch(Uim + GIDX(2, sX), 0, 3);
  __builtin_prefetch(Ure + GIDX(3, sT), 0, 3);
  __builtin_prefetch(Uim + GIDX(3, sT), 0, 3);
  __builtin_prefetch(Ure + GIDX(3, sX), 0, 3);
  __builtin_prefetch(Uim + GIDX(3, sX), 0, 3);

  asm volatile("s_wait_asynccnt 0x0" ::: "memory");
  __syncthreads();

#define PLAQ(pairIdx)                                                     \
  do {                                                                    \
    mmul(T1, F1, F2);                                                     \
    mmul_dag(T2, T1, F3);                                                 \
    mmul_dag(T1, T2, F4);                                                 \
    store_mat(out + ((size_t)(pairIdx) * VSITES + s) * 18, T1);           \
  } while (0)

  // pair 0: (mu=1, nu=0)   F2/F3 already loaded from global above
  lmat(F1, lds, MI(1, yl, z));
  lmat(F4, lds, MI(0, yl, z));
  PLAQ(0);

  // pair 1: (mu=2, nu=0)
  lmat(F1, lds, MI(2, yl, z));
  lmat(F2, lds, MI(0, yl + 1, z));          // U_0(x+e_y) from halo row
  gmat(F3, Ure, Uim, GIDX(2, sT));
  lmat(F4, lds, MI(0, yl, z));
  PLAQ(1);

  // pair 2: (mu=2, nu=1)
  lmat(F1, lds, MI(2, yl, z));
  lmat(F2, lds, MI(1, yl + 1, z));
  gmat(F3, Ure, Uim, GIDX(2, sX));
  lmat(F4, lds, MI(1, yl, z));
  PLAQ(2);

  // pair 3: (mu=3, nu=0)
  lmat(F1, lds, MI(3, yl, z));
  lmat(F2, lds, MI(0, yl, zp));             // U_0(x+e_z), periodic inside z-line
  gmat(F3, Ure, Uim, GIDX(3, sT));
  lmat(F4, lds, MI(0, yl, z));
  PLAQ(3);

  // pair 4: (mu=3, nu=1)
  lmat(F1, lds, MI(3, yl, z));
  lmat(F2, lds, MI(1, yl, zp));
  gmat(F3, Ure, Uim, GIDX(3, sX));
  lmat(F4, lds, MI(1, yl, z));
  PLAQ(4);

  // pair 5: (mu=3, nu=2)
  lmat(F1, lds, MI(3, yl, z));
  lmat(F2, lds, MI(2, yl, zp));
  lmat(F3, lds, MI(3, yl + 1, z));          // U_3(x+e_y) from halo row
  lmat(F4, lds, MI(2, yl, z));
  PLAQ(5);

#undef PLAQ
#undef GIDX
#undef MI
}

extern "C" void kernel_launch(void* const* d_in, const int* in_sizes, int n_in,
                              void* d_out, int out_size, void* d_ws, size_t ws_size,
                              hipStream_t stream) {
  const float* Ure = (const float*)d_in[0];
  const float* Uim = (const float*)d_in[1];
  float* out = (float*)d_out;
  const int nblocks = T_DIM * X_DIM * (Y_DIM / YTILE);   // 3456
  plaquette_kernel<<<nblocks, NTHREADS, 0, stream>>>(Ure, Uim, out);
}